// Attention3_70841190580882
// MI455X (gfx1250) — compile-verified
//
#include <hip/hip_runtime.h>

// ---------------------------------------------------------------------------
// Attention3: q=enc@Wq, k=dec@Wk, v=lat@Wv, S=q k^T/sqrt(H) masked on q-rows,
// softmax over the QUERY axis (per key column), out = softmax(S) @ v.
// All matmuls use CDNA5 V_WMMA_F32_16X16X4_F32 (exact fp32, wave32).
// GEMM is software-pipelined: global->reg prefetch of tile i+1 overlaps the
// WMMAs of tile i via double-buffered LDS (one barrier per K-step), plus
// global_prefetch_b8 two tiles ahead.
// ---------------------------------------------------------------------------

typedef __attribute__((ext_vector_type(2))) float v2f;
typedef __attribute__((ext_vector_type(4))) float f32x4;
typedef __attribute__((ext_vector_type(8))) float v8f;

#define WMMA_F32(a, b, c) \
  __builtin_amdgcn_wmma_f32_16x16x4_f32(false, (a), false, (b), (short)0, (c), false, false)

// ---------------------------------------------------------------------------
// Generic fp32 WMMA GEMM: C[M,N] = scale * (A[M,K] x B), optional q-row mask.
//   B_SRC_NK = true : B source indexed [n][k] (k contiguous)  -> direct copy
//   B_SRC_NK = false: B source indexed [k][n] (n contiguous)  -> transpose stage
// Workgroup: 256 threads (8 wave32), tile 128(M) x 64(N), K staged 32 wide.
// Each wave computes a 32x32 sub-tile as 2x2 WMMA 16x16 frags.
// M, N, K are exact multiples of the tile sizes for every launch below.
// ---------------------------------------------------------------------------
template <bool B_SRC_NK, bool MASKED>
__global__ __launch_bounds__(256) void gemm_wmma_f32(
    const float* __restrict__ Ag, const float* __restrict__ Bg,
    float* __restrict__ Cg, int K, int lda, int ldb, int ldc,
    long long sA, long long sB, long long sC,
    float scale, const int* __restrict__ maskp, int sMask) {
  constexpr int TM = 128, TN = 64, TK = 32, LDT = TK + 4;  // 144B rows, 16B aligned
  __shared__ __align__(16) float As[2][TM][LDT];  // 36 KB
  __shared__ __align__(16) float Bs[2][TN][LDT];  // 18 KB

  const int t = threadIdx.x;
  const int z = blockIdx.z;
  const int tileM = blockIdx.y * TM;
  const int tileN = blockIdx.x * TN;

  const float* A = Ag + (size_t)z * (size_t)sA;
  const float* B = Bg + (size_t)z * (size_t)sB;
  float* C = Cg + (size_t)z * (size_t)sC;

  const int wave = t >> 5, lane = t & 31;
  const int wm = (wave & 3) * 32;   // 4 wave-rows  * 32 = 128
  const int wn = (wave >> 2) * 32;  // 2 wave-cols  * 32 = 64
  const int lrow = lane & 15;       // row/col within 16x16 frag
  const int lhi = lane >> 4;        // lane-half selects K pair / M half

  v8f acc[2][2] = {};

  // staging decompositions (256 threads)
  const int ar = t >> 1, ac = (t & 1) * 16;  // A: 128 rows x 32 f32 (64B/thread)
  const int bn = t >> 2, bs4 = (t & 3) * 8;  // B [n][k]: 64 rows x 32 f32
  const int bk = t >> 3, bn8 = (t & 7) * 8;  // B [k][n]: 32 k  x 64 n (transpose)

  // ---- global -> register staging (software pipeline) ----
  f32x4 ra0, ra1, ra2, ra3;  // A: 16 floats/thread
  f32x4 rb0, rb1;            // B:  8 floats/thread

  auto loadA = [&](int k0) {
    const float* s = A + (size_t)(tileM + ar) * lda + (k0 + ac);
    ra0 = *(const f32x4*)(s + 0);
    ra1 = *(const f32x4*)(s + 4);
    ra2 = *(const f32x4*)(s + 8);
    ra3 = *(const f32x4*)(s + 12);
  };
  auto loadB = [&](int k0) {
    if (B_SRC_NK) {
      const float* s = B + (size_t)(tileN + bn) * ldb + (k0 + bs4);
      rb0 = *(const f32x4*)(s + 0);
      rb1 = *(const f32x4*)(s + 4);
    } else {
      const float* s = B + (size_t)(k0 + bk) * ldb + (tileN + bn8);
      rb0 = *(const f32x4*)(s + 0);
      rb1 = *(const f32x4*)(s + 4);
    }
  };
  auto storeTile = [&](int buf) {
    f32x4* d = (f32x4*)&As[buf][ar][ac];
    d[0] = ra0; d[1] = ra1; d[2] = ra2; d[3] = ra3;
    if (B_SRC_NK) {
      f32x4* e = (f32x4*)&Bs[buf][bn][bs4];
      e[0] = rb0; e[1] = rb1;
    } else {  // transpose into Bs[n][k]
      Bs[buf][bn8 + 0][bk] = rb0[0]; Bs[buf][bn8 + 1][bk] = rb0[1];
      Bs[buf][bn8 + 2][bk] = rb0[2]; Bs[buf][bn8 + 3][bk] = rb0[3];
      Bs[buf][bn8 + 4][bk] = rb1[0]; Bs[buf][bn8 + 5][bk] = rb1[1];
      Bs[buf][bn8 + 6][bk] = rb1[2]; Bs[buf][bn8 + 7][bk] = rb1[3];
    }
  };

  loadA(0);
  loadB(0);
  int buf = 0;

  for (int k0 = 0; k0 < K; k0 += TK) {
    storeTile(buf);
    __syncthreads();  // tile `buf` visible; also guarantees prior readers of
                      // the buffer we will overwrite NEXT iteration are done.

    if (k0 + TK < K) {  // issue next tile's global loads before computing
      loadA(k0 + TK);
      loadB(k0 + TK);
      if (k0 + 2 * TK < K) {  // pull tile i+2 toward cache (global_prefetch_b8)
        __builtin_prefetch(A + (size_t)(tileM + ar) * lda + (k0 + 2 * TK + ac), 0, 1);
        if (B_SRC_NK)
          __builtin_prefetch(B + (size_t)(tileN + bn) * ldb + (k0 + 2 * TK + bs4), 0, 1);
        else
          __builtin_prefetch(B + (size_t)(k0 + 2 * TK + bk) * ldb + (tileN + bn8), 0, 1);
      }
    }

    // ISA layout for V_WMMA_F32_16X16X4_F32:
    //   A 16x4: lanes 0-15 row M=lane hold {K0,K1}; lanes 16-31 hold {K2,K3}
    //   B 4x16: lanes 0-15 col N=lane hold {K0,K1}; lanes 16-31 hold {K2,K3}
    const float* a0p = &As[buf][wm + lrow][2 * lhi];
    const float* a1p = &As[buf][wm + 16 + lrow][2 * lhi];
    const float* b0p = &Bs[buf][wn + lrow][2 * lhi];
    const float* b1p = &Bs[buf][wn + 16 + lrow][2 * lhi];
#pragma unroll
    for (int kk = 0; kk < TK; kk += 4) {
      v2f a0 = *(const v2f*)(a0p + kk);
      v2f a1 = *(const v2f*)(a1p + kk);
      v2f b0 = *(const v2f*)(b0p + kk);
      v2f b1 = *(const v2f*)(b1p + kk);
      acc[0][0] = WMMA_F32(a0, b0, acc[0][0]);
      acc[0][1] = WMMA_F32(a0, b1, acc[0][1]);
      acc[1][0] = WMMA_F32(a1, b0, acc[1][0]);
      acc[1][1] = WMMA_F32(a1, b1, acc[1][1]);
    }
    buf ^= 1;
  }

  // C/D layout: VGPR i -> M = i + 8*(lane/16), N = lane%16
  const int* mrow = MASKED ? (maskp + (size_t)z * sMask) : (const int*)nullptr;
#pragma unroll
  for (int si = 0; si < 2; ++si) {
#pragma unroll
    for (int ti = 0; ti < 2; ++ti) {
      const int r0 = tileM + wm + si * 16 + lhi * 8;
      const int c = tileN + wn + ti * 16 + lrow;
#pragma unroll
      for (int i = 0; i < 8; ++i) {
        float v = acc[si][ti][i] * scale;
        if (MASKED) {
          if (mrow[r0 + i] == 0) v = -1.0e9f;  // mask over query rows
        }
        C[(size_t)(r0 + i) * ldc + c] = v;
      }
    }
  }
}

// ---------------------------------------------------------------------------
// Softmax over the QUERY axis: per (b, key-col) online (max, sumexp) over q.
// Split q into gridDim.z chunks for parallelism; combine afterwards.
// ---------------------------------------------------------------------------
__device__ __forceinline__ void online_upd(float& m, float& s, float v) {
  float nm = fmaxf(m, v);
  s = s * __expf(m - nm) + __expf(v - nm);
  m = nm;
}

__global__ __launch_bounds__(256) void col_stats_partial(
    const float* __restrict__ S, float* __restrict__ pm, float* __restrict__ ps,
    int L, int qChunk) {
  const int col = blockIdx.x * 256 + threadIdx.x;
  const int b = blockIdx.y;
  const int part = blockIdx.z;
  const float* p = S + (size_t)b * L * L + (size_t)part * qChunk * L + col;
  float m0 = -INFINITY, m1 = -INFINITY, m2 = -INFINITY, m3 = -INFINITY;
  float s0 = 0.f, s1 = 0.f, s2 = 0.f, s3 = 0.f;
  for (int q = 0; q < qChunk; q += 4) {  // 4 independent chains for MLP
    float v0 = p[(size_t)(q + 0) * L];
    float v1 = p[(size_t)(q + 1) * L];
    float v2 = p[(size_t)(q + 2) * L];
    float v3 = p[(size_t)(q + 3) * L];
    online_upd(m0, s0, v0);
    online_upd(m1, s1, v1);
    online_upd(m2, s2, v2);
    online_upd(m3, s3, v3);
  }
  float M = fmaxf(fmaxf(m0, m1), fmaxf(m2, m3));
  float Sm = s0 * __expf(m0 - M) + s1 * __expf(m1 - M) +
             s2 * __expf(m2 - M) + s3 * __expf(m3 - M);
  const size_t n = (size_t)gridDim.y * L;  // B*L
  const size_t o = (size_t)part * n + (size_t)b * L + col;
  pm[o] = M;
  ps[o] = Sm;
}

__global__ __launch_bounds__(256) void col_stats_combine(
    const float* __restrict__ pm, const float* __restrict__ ps,
    float* __restrict__ cmax, float* __restrict__ csum, int n, int parts) {
  const int i = blockIdx.x * 256 + threadIdx.x;
  if (i >= n) return;
  float M = -INFINITY, S = 0.f;
  for (int p = 0; p < parts; ++p) {
    float m = pm[(size_t)p * n + i];
    float s = ps[(size_t)p * n + i];
    float nm = fmaxf(M, m);
    S = S * __expf(M - nm) + s * __expf(m - nm);
    M = nm;
  }
  cmax[i] = M;
  csum[i] = S;
}

__global__ __launch_bounds__(256) void softmax_normalize(
    float* __restrict__ S, const float* __restrict__ cmax,
    const float* __restrict__ csum, int L) {
  const int b = blockIdx.y;
  const size_t i = (size_t)blockIdx.x * 256 + threadIdx.x;  // index within [L*L)
  const int col = (int)(i % (size_t)L);
  const float m = cmax[(size_t)b * L + col];
  const float rs = 1.0f / csum[(size_t)b * L + col];
  const size_t idx = (size_t)b * L * L + i;
  S[idx] = __expf(S[idx] - m) * rs;
}

// ---------------------------------------------------------------------------
extern "C" void kernel_launch(void* const* d_in, const int* in_sizes, int n_in,
                              void* d_out, int out_size, void* d_ws, size_t ws_size,
                              hipStream_t stream) {
  (void)in_sizes; (void)n_in; (void)out_size; (void)ws_size;
  constexpr int B = 32, L = 1024, H = 1024, Z = 512;
  constexpr int PARTS = 8;

  const float* enc = (const float*)d_in[0];
  const float* dec = (const float*)d_in[1];
  const float* lat = (const float*)d_in[2];
  const int* msk = (const int*)d_in[3];
  const float* Wq = (const float*)d_in[4];
  const float* Wk = (const float*)d_in[5];
  const float* Wv = (const float*)d_in[6];

  float* out = (float*)d_out;                 // [B,L,Z]
  float* attn = out + (size_t)B * L * Z;      // [B,L,L] (scores -> softmaxed in place)

  float* qf = (float*)d_ws;                   // [B*L, H] fp32
  float* kf = qf + (size_t)B * L * H;         // [B*L, H]
  float* vf = kf + (size_t)B * L * H;         // [B*L, Z]
  float* cmax = vf + (size_t)B * L * Z;       // [B*L]
  float* csum = cmax + (size_t)B * L;         // [B*L]
  float* pm = csum + (size_t)B * L;           // [PARTS, B*L]
  float* ps = pm + (size_t)PARTS * B * L;     // [PARTS, B*L]

  const dim3 blk(256);

  // 1-3) q/k/v projections: batch folded into M (weights shared). C = fp32 ws.
  gemm_wmma_f32<false, false><<<dim3(H / 64, (B * L) / 128, 1), blk, 0, stream>>>(
      enc, Wq, qf, H, H, H, H, 0, 0, 0, 1.0f, nullptr, 0);
  gemm_wmma_f32<false, false><<<dim3(H / 64, (B * L) / 128, 1), blk, 0, stream>>>(
      dec, Wk, kf, H, H, H, H, 0, 0, 0, 1.0f, nullptr, 0);
  gemm_wmma_f32<false, false><<<dim3(Z / 64, (B * L) / 128, 1), blk, 0, stream>>>(
      lat, Wv, vf, Z, Z, Z, Z, 0, 0, 0, 1.0f, nullptr, 0);

  // 4) raw scores: S[b,q,k] = (q_b . k_b)/32, masked rows -> -1e9
  gemm_wmma_f32<true, true><<<dim3(L / 64, L / 128, B), blk, 0, stream>>>(
      qf, kf, attn, H, H, H, L,
      (long long)L * H, (long long)L * H, (long long)L * L,
      0.03125f /* 1/sqrt(1024) */, msk, L);

  // 5) softmax over query axis: per-(b,key) stats, then normalize in place
  col_stats_partial<<<dim3(L / 256, B, PARTS), blk, 0, stream>>>(attn, pm, ps, L, L / PARTS);
  col_stats_combine<<<dim3((B * L) / 256), blk, 0, stream>>>(pm, ps, cmax, csum, B * L, PARTS);
  softmax_normalize<<<dim3((L * L) / 256, B), blk, 0, stream>>>(attn, cmax, csum, L);

  // 6) out = attn @ v
  gemm_wmma_f32<false, false><<<dim3(Z / 64, L / 128, B), blk, 0, stream>>>(
      attn, vf, out, L, L, Z, Z,
      (long long)L * L, (long long)L * Z, (long long)L * Z, 1.0f, nullptr, 0);
}